// FilterDetections_90348932038752
// MI455X (gfx1250) — compile-verified
//
#include <hip/hip_runtime.h>
#include <stdint.h>

// Problem constants (from the reference)
#define BB        2
#define NN        4096
#define CC        8
#define MAXDET    100
#define SCORE_THR 0.01f
#define NMS_THR   0.5f
#define NEGV      (-1e30f)
#define THREADS   1024

// ---------------------------------------------------------------------------
// CDNA5 async DMA: global -> LDS, tracked by ASYNCcnt (CDNA5 ISA)
// GVS mode: mem_addr = SADDR64 + VADDR32 (+ inst offset); VDST holds the
// wave-relative LDS byte address. 16 bytes per lane per issue.
// ---------------------------------------------------------------------------
__device__ __forceinline__ void async_load_b128(uint32_t lds_byte_addr,
                                                uint32_t global_byte_off,
                                                const void* global_base) {
  asm volatile("global_load_async_to_lds_b128 %0, %1, %2"
               :
               : "v"(lds_byte_addr), "v"(global_byte_off), "s"(global_base)
               : "memory");
}

__device__ __forceinline__ void wait_async0() {
#if __has_builtin(__builtin_amdgcn_s_wait_asynccnt)
  __builtin_amdgcn_s_wait_asynccnt(0);
#else
  asm volatile("s_wait_asynccnt 0" ::: "memory");
#endif
}

__device__ __forceinline__ uint32_t lds_offset_of(const void* p) {
  // Generic pointers to LDS carry the wave-relative LDS byte offset in the
  // low 32 bits (aperture bits live in the high word) — ISA §10.2.
  return (uint32_t)(uintptr_t)p;
}

// ---------------------------------------------------------------------------
// Kernel 1: one workgroup per (batch, class). Sort + greedy NMS, all in LDS.
// Writes masked scores (kept ? score : NEG) to ws as [B][C][N].
// ---------------------------------------------------------------------------
extern "C" __global__ void __launch_bounds__(THREADS)
nms_per_class_kernel(const float* __restrict__ boxes,  // [B][N][4]
                     const float* __restrict__ cls,    // [B][N][C]
                     float* __restrict__ masked)       // [B][C][N]
{
  extern __shared__ char smem[];
  float*    s_box   = (float*)(smem);                     // 65536 B (orig order)
  float*    s_score = (float*)(smem + 65536);             // 16384 B (sorted)
  int*      s_idx   = (int*)  (smem + 65536 + 16384);     // 16384 B (sorted -> orig)
  uint32_t* s_sup   = (uint32_t*)(smem + 98304);          //   512 B suppression bits
  int*      s_cnt   = (int*)  (smem + 98816);             //     4 B valid count

  const int tid = threadIdx.x;
  const int b   = blockIdx.x / CC;
  const int c   = blockIdx.x % CC;

  const float* gbox = boxes + (size_t)b * NN * 4;

  // ---- async DMA: 64 KB of boxes into LDS (4 x b128 per lane) ----
  const uint32_t lds_box = lds_offset_of(s_box);
  #pragma unroll
  for (int r = 0; r < 4; ++r) {
    uint32_t byteoff = (uint32_t)(tid + r * THREADS) * 16u;
    async_load_b128(lds_box + byteoff, byteoff, (const void*)gbox);
  }

  if (tid == 0) *s_cnt = 0;
  __syncthreads();

  // ---- class scores: mask invalid to NEG (matches where(valid, s, NEG)) ----
  __builtin_prefetch(cls + ((size_t)b * NN) * CC, 0, 0);  // global_prefetch_b8
  int lcnt = 0;
  for (int i = tid; i < NN; i += THREADS) {
    float sc   = cls[((size_t)b * NN + i) * CC + c];
    bool valid = sc > SCORE_THR;
    s_score[i] = valid ? sc : NEGV;
    s_idx[i]   = i;
    lcnt += valid ? 1 : 0;
  }
  atomicAdd(s_cnt, lcnt);
  for (int w = tid; w < NN / 32; w += THREADS) s_sup[w] = 0u;

  wait_async0();     // drain this wave's ASYNCcnt before anyone reads s_box
  __syncthreads();

  const int M = *s_cnt;   // #valid; sorted-desc puts them first (NEG sinks)

  // ---- bitonic sort: score desc, tie -> lower original index (stable-ish,
  //      matches jnp.argsort(-masked) tie behavior) ----
  for (int k = 2; k <= NN; k <<= 1) {
    for (int j = k >> 1; j > 0; j >>= 1) {
      __syncthreads();
      for (int i = tid; i < NN; i += THREADS) {
        int p = i ^ j;
        if (p > i) {
          float si = s_score[i], sp = s_score[p];
          int   ii = s_idx[i],   ip = s_idx[p];
          bool before = (si > sp) || (si == sp && ii < ip); // i precedes p (desc)
          bool descBlock = ((i & k) == 0);
          bool doSwap = descBlock ? !before : before;
          if (doSwap) {
            s_score[i] = sp; s_score[p] = si;
            s_idx[i]   = ip; s_idx[p]   = ii;
          }
        }
      }
    }
  }
  __syncthreads();

  // ---- greedy NMS over sorted order; IoU on the fly from LDS boxes ----
  for (int i = 0; i < M; ++i) {
    // Uniform read: previous iteration's atomicOrs are fenced by the barrier.
    bool kept = ((s_sup[i >> 5] >> (i & 31)) & 1u) == 0u;
    if (kept) {
      const int   oi = s_idx[i];
      const float x1 = s_box[oi * 4 + 0], y1 = s_box[oi * 4 + 1];
      const float x2 = s_box[oi * 4 + 2], y2 = s_box[oi * 4 + 3];
      const float ai = (x2 - x1) * (y2 - y1);
      for (int jj = i + 1 + tid; jj < M; jj += THREADS) {
        const int   oj  = s_idx[jj];
        const float bx1 = s_box[oj * 4 + 0], by1 = s_box[oj * 4 + 1];
        const float bx2 = s_box[oj * 4 + 2], by2 = s_box[oj * 4 + 3];
        const float aj  = (bx2 - bx1) * (by2 - by1);
        const float ix1 = fmaxf(x1, bx1), iy1 = fmaxf(y1, by1);
        const float ix2 = fminf(x2, bx2), iy2 = fminf(y2, by2);
        const float iw  = fmaxf(ix2 - ix1, 0.0f);
        const float ih  = fmaxf(iy2 - iy1, 0.0f);
        const float inter = iw * ih;
        const float uni   = fmaxf(ai + aj - inter, 1e-9f);
        if (inter > NMS_THR * uni)                 // iou > 0.5
          atomicOr(&s_sup[jj >> 5], 1u << (jj & 31));
      }
    }
    __syncthreads();
  }

  // ---- scatter masked scores back to original order in ws ----
  float* mrow = masked + ((size_t)(b * CC + c)) * NN;
  for (int i = tid; i < NN; i += THREADS) {
    bool kept = (i < M) && (((s_sup[i >> 5] >> (i & 31)) & 1u) == 0u);
    mrow[s_idx[i]] = kept ? s_score[i] : NEGV;
  }
}

// ---------------------------------------------------------------------------
// Kernel 2: one workgroup per image. Top-100 over the C*N masked matrix
// (LDS-resident, 128 KB), then gather + write all outputs.
// Output layout (flat, return order):
//   boxes  [B][100][4]  @ 0
//   scores [B][100]     @ 800
//   labels [B][100]     @ 1000
//   rot    [B][100][3]  @ 1200
//   trans  [B][100][3]  @ 1800        total 2400 floats
// ---------------------------------------------------------------------------
extern "C" __global__ void __launch_bounds__(THREADS)
topk_gather_kernel(const float* __restrict__ boxes,   // [B][N][4]
                   const float* __restrict__ rot,     // [B][N][3]
                   const float* __restrict__ trans,   // [B][N][3]
                   const float* __restrict__ masked,  // [B][C][N]
                   float* __restrict__ out)
{
  extern __shared__ char smem[];
  float* s_val  = (float*)(smem);             // 131072 B : C*N masked scores
  float* s_rv   = (float*)(smem + 131072);    //   4096 B : reduction values
  int*   s_ri   = (int*)  (smem + 135168);    //   4096 B : reduction indices
  float* s_topv = (float*)(smem + 139264);    //    400 B
  int*   s_topi = (int*)  (smem + 139664);    //    400 B

  const int tid = threadIdx.x;
  const int b   = blockIdx.x;
  const float* src = masked + (size_t)b * CC * NN;

  // ---- async DMA: 128 KB of masked scores into LDS (8 x b128 per lane) ----
  const uint32_t lds_val = lds_offset_of(s_val);
  #pragma unroll
  for (int r = 0; r < 8; ++r) {
    uint32_t byteoff = (uint32_t)(tid + r * THREADS) * 16u;
    async_load_b128(lds_val + byteoff, byteoff, (const void*)src);
  }
  wait_async0();
  __syncthreads();

  // ---- 100 successive argmax passes (tie -> lower flat index, = lax.top_k) ----
  for (int sel = 0; sel < MAXDET; ++sel) {
    float bv = NEGV - 1.0f;   // strictly below any stored value incl. NEGV? NEGV-1==NEGV in fp; handle via index
    int   bi = CC * NN;       // sentinel: larger than any real index
    for (int f = tid; f < CC * NN; f += THREADS) {
      float v = s_val[f];
      if (v > bv || (v == bv && f < bi)) { bv = v; bi = f; }
    }
    s_rv[tid] = bv; s_ri[tid] = bi;
    __syncthreads();
    for (int off = THREADS / 2; off > 0; off >>= 1) {
      if (tid < off) {
        float ov = s_rv[tid + off]; int oi = s_ri[tid + off];
        float mv = s_rv[tid];       int mi = s_ri[tid];
        if (ov > mv || (ov == mv && oi < mi)) { s_rv[tid] = ov; s_ri[tid] = oi; }
      }
      __syncthreads();
    }
    if (tid == 0) {
      s_topv[sel] = s_rv[0];
      s_topi[sel] = s_ri[0];
      if (s_ri[0] < CC * NN) s_val[s_ri[0]] = NEGV;  // remove from next passes
    }
    __syncthreads();
  }

  // ---- gather winners and write all five outputs ----
  const float* gbox = boxes + (size_t)b * NN * 4;
  const float* grot = rot   + (size_t)b * NN * 3;
  const float* gtr  = trans + (size_t)b * NN * 3;

  for (int r = tid; r < MAXDET; r += THREADS) {
    const float v   = s_topv[r];
    const int   fi  = (s_topi[r] < CC * NN) ? s_topi[r] : 0;
    const int   lab = fi >> 12;        // / NN
    const int   anc = fi & (NN - 1);   // % NN
    const bool  ok  = v > (NEGV * 0.5f);

    float* obox = out + ((size_t)b * MAXDET + r) * 4;
    #pragma unroll
    for (int k = 0; k < 4; ++k)
      obox[k] = ok ? gbox[anc * 4 + k] : -1.0f;

    out[BB * MAXDET * 4 + b * MAXDET + r]                 = ok ? v : -1.0f;        // scores @800
    out[BB * MAXDET * 5 + b * MAXDET + r]                 = ok ? (float)lab : -1.0f; // labels @1000

    float* orot = out + BB * MAXDET * 6 + ((size_t)b * MAXDET + r) * 3;            // rot @1200
    float* otr  = out + BB * MAXDET * 9 + ((size_t)b * MAXDET + r) * 3;            // trans @1800
    #pragma unroll
    for (int k = 0; k < 3; ++k) {
      orot[k] = ok ? grot[anc * 3 + k] : -1.0f;
      otr[k]  = ok ? gtr[anc * 3 + k]  : -1.0f;
    }
  }
}

// ---------------------------------------------------------------------------
extern "C" void kernel_launch(void* const* d_in, const int* in_sizes, int n_in,
                              void* d_out, int out_size, void* d_ws, size_t ws_size,
                              hipStream_t stream) {
  (void)in_sizes; (void)n_in; (void)out_size; (void)ws_size;
  const float* boxes = (const float*)d_in[0];
  const float* cls   = (const float*)d_in[1];
  const float* rot   = (const float*)d_in[2];
  const float* trans = (const float*)d_in[3];
  float* out    = (float*)d_out;
  float* masked = (float*)d_ws;   // B*C*N floats = 256 KB scratch

  const size_t smem1 = 65536 + 16384 + 16384 + 512 + 64;            // ~98.9 KB
  const size_t smem2 = 131072 + 4096 + 4096 + 400 + 400 + 32;       // ~140.0 KB

  hipLaunchKernelGGL(nms_per_class_kernel, dim3(BB * CC), dim3(THREADS),
                     smem1, stream, boxes, cls, masked);
  hipLaunchKernelGGL(topk_gather_kernel, dim3(BB), dim3(THREADS),
                     smem2, stream, boxes, rot, trans, masked, out);
}